// SimpleSSMCell_87351044866245
// MI455X (gfx1250) — compile-verified
//
#include <hip/hip_runtime.h>

typedef __attribute__((ext_vector_type(2))) float v2f;
typedef __attribute__((ext_vector_type(8))) float v8f;

#define MTOT   16384   // B*T
#define KDIM   1024
#define NDIM   1024
#define TSEQ   2048
#define BATCH  8
#define NCHUNK 16
#define CHUNKLEN 128   // TSEQ / NCHUNK
#define LDSTR  68      // padded LDS row stride (272B: 16B-aligned, conflict-light)
#define KTILE  64
#define NKT    (KDIM / KTILE)   // 16 K-tiles

// ---- CDNA5 async global->LDS copy (ASYNCcnt-tracked, bypasses VGPRs) -----
__device__ __forceinline__ void async_copy_b128(unsigned lds_byte_off,
                                                const float* gaddr) {
    asm volatile("global_load_async_to_lds_b128 %0, %1, off"
                 :: "v"(lds_byte_off), "v"(gaddr) : "memory");
}
__device__ __forceinline__ unsigned lds_off(const void* p) {
    // generic (flat) pointer to __shared__: low 32 bits == LDS byte offset
    return (unsigned)(unsigned long long)p;
}

// out[m][n] = scale * ( A1[m,:]·W1[n,:] + bias1[n] (+ A2[m,:]·W2[n,:] + bias2[n]) )
// scale = exp(log_dt[0]) if log_dt != null else 1.0
__launch_bounds__(256, 2)
__global__ void ssm_gemm_wmma(const float* __restrict__ A1,
                              const float* __restrict__ W1,
                              const float* __restrict__ bias1,
                              const float* __restrict__ A2,
                              const float* __restrict__ W2,
                              const float* __restrict__ bias2,
                              const float* __restrict__ log_dt,
                              float* __restrict__ out)
{
    __shared__ float lds_a[2][64 * LDSTR];    // ~17.4 KB x2
    __shared__ float lds_w[2][128 * LDSTR];   // ~34.8 KB x2

    const int tid  = threadIdx.x;
    const int wv   = tid >> 5;       // wave 0..7 -> 16-col slice of the N-tile
    const int lane = tid & 31;
    const int hlf  = lane >> 4;      // 0: lanes 0-15, 1: lanes 16-31
    const int ln   = lane & 15;

    const int m0 = blockIdx.y * 64;   // 4 x 16-row subtiles per wave
    const int n0 = blockIdx.x * 128;  // 8 waves x 16 cols

    v8f acc[4];
#pragma unroll
    for (int mi = 0; mi < 4; ++mi) acc[mi] = (v8f){0.f,0.f,0.f,0.f,0.f,0.f,0.f,0.f};

    // per-thread fill coordinates (A: 4 x float4, W: 8 x float4 per tile)
    const int fa_row = tid >> 4;            // +16 per step (4 steps -> 64 rows)
    const int fa_c4  = (tid & 15) << 2;

    for (int pass = 0; pass < 2; ++pass) {
        const float* Ap = pass ? A2 : A1;
        const float* Wp = pass ? W2 : W1;
        if (!Ap) break;

        // issue async fills of one K-tile into LDS buffer `buf`
        auto issue_fill = [&](int buf, int k0) {
#pragma unroll
            for (int i = 0; i < 4; ++i) {            // A tile: 64 x 64
                const int row = fa_row + (i << 4);
                async_copy_b128(lds_off(&lds_a[buf][row * LDSTR + fa_c4]),
                                Ap + (size_t)(m0 + row) * KDIM + k0 + fa_c4);
            }
#pragma unroll
            for (int i = 0; i < 8; ++i) {            // W tile: 128 x 64
                const int idx = tid + (i << 8);
                const int row = idx >> 4;
                const int c4  = (idx & 15) << 2;
                async_copy_b128(lds_off(&lds_w[buf][row * LDSTR + c4]),
                                Wp + (size_t)(n0 + row) * KDIM + k0 + c4);
            }
        };

        issue_fill(0, 0);                            // prologue

        for (int jt = 0; jt < NKT; ++jt) {
            const int  cur     = jt & 1;
            const bool hasnext = (jt + 1) < NKT;
            if (hasnext) {
                issue_fill(cur ^ 1, (jt + 1) * KTILE);
                // 12 next-tile copies in flight; current tile (issued earlier,
                // completes in order) is done once outstanding <= 12
                asm volatile("s_wait_asynccnt 0xc" ::: "memory");
            } else {
                asm volatile("s_wait_asynccnt 0x0" ::: "memory");
            }
            __syncthreads();                         // cur buffer visible to all

            // ---- 64 x V_WMMA_F32_16X16X4_F32 per wave over this K-tile
            // A frag (16x4): lane ln = row; VGPR0/1 = K = 2*hlf, 2*hlf+1
            // B frag (4x16): lane ln = col; VGPR0/1 = K = 2*hlf, 2*hlf+1
            const float* ap = &lds_a[cur][ln * LDSTR];
            const float* wp = &lds_w[cur][(wv * 16 + ln) * LDSTR];
#pragma unroll
            for (int kk = 0; kk < KTILE; kk += 4) {
                const int ko = kk + (hlf << 1);
                const v2f b = *(const v2f*)(wp + ko);
#pragma unroll
                for (int mi = 0; mi < 4; ++mi) {
                    const v2f a = *(const v2f*)(ap + (mi << 4) * LDSTR + ko);
                    acc[mi] = __builtin_amdgcn_wmma_f32_16x16x4_f32(
                        false, a, false, b, (short)0, acc[mi], false, false);
                }
            }
            __syncthreads();         // compute done before cur is refilled
        }
    }

    float scale = 1.0f;
    if (log_dt) scale = __expf(log_dt[0]);
    const int col = n0 + wv * 16 + ln;
    const float bias = bias1[col] + (bias2 ? bias2[col] : 0.0f);

    // C/D layout: VGPR r, lanes 0-15 -> M = r, lanes 16-31 -> M = r + 8; N = ln
#pragma unroll
    for (int mi = 0; mi < 4; ++mi) {
#pragma unroll
        for (int r = 0; r < 8; ++r) {
            const int m = m0 + (mi << 4) + r + (hlf << 3);
            out[(size_t)m * NDIM + col] = scale * (acc[mi][r] + bias);
        }
    }
}

// ---- chunked diagonal scan: x_t = abar * x_{t-1} + Bu_t ------------------

__global__ void ssm_scan_pass1(const float* __restrict__ Bu,
                               const float* __restrict__ log_A,
                               const float* __restrict__ log_dt,
                               float* __restrict__ chunk_state)
{
    const int idx = blockIdx.x * blockDim.x + threadIdx.x; // [0, B*NCHUNK*N)
    const int n = idx & (NDIM - 1);
    const int c = (idx >> 10) & (NCHUNK - 1);
    const int b = idx >> 14;
    const float dt   = __expf(log_dt[0]);
    const float abar = __expf(-__expf(log_A[n]) * dt);
    const float* p = Bu + ((size_t)b * TSEQ + (size_t)c * CHUNKLEN) * NDIM + n;
    float x = 0.f;
#pragma unroll 4
    for (int t = 0; t < CHUNKLEN; ++t)
        x = fmaf(abar, x, p[(size_t)t * NDIM]);
    chunk_state[((size_t)c * BATCH + b) * NDIM + n] = x;
}

__global__ void ssm_scan_pass2(float* __restrict__ chunk_state,
                               const float* __restrict__ log_A,
                               const float* __restrict__ log_dt)
{
    const int idx = blockIdx.x * blockDim.x + threadIdx.x; // [0, B*N)
    const int n = idx & (NDIM - 1);
    const int b = idx >> 10;
    const float dt = __expf(log_dt[0]);
    const float abar_ck = __expf(-__expf(log_A[n]) * dt * (float)CHUNKLEN); // abar^CHUNKLEN
    float carry = 0.f;
    for (int c = 0; c < NCHUNK; ++c) {
        const size_t o = ((size_t)c * BATCH + b) * NDIM + n;
        const float s = chunk_state[o];
        chunk_state[o] = carry;                 // exclusive carry-in for chunk c
        carry = fmaf(abar_ck, carry, s);
    }
}

__global__ void ssm_scan_pass3(float* __restrict__ Bu,   // overwritten with xs
                               const float* __restrict__ chunk_state,
                               const float* __restrict__ log_A,
                               const float* __restrict__ log_dt)
{
    const int idx = blockIdx.x * blockDim.x + threadIdx.x;
    const int n = idx & (NDIM - 1);
    const int c = (idx >> 10) & (NCHUNK - 1);
    const int b = idx >> 14;
    const float dt   = __expf(log_dt[0]);
    const float abar = __expf(-__expf(log_A[n]) * dt);
    float x = chunk_state[((size_t)c * BATCH + b) * NDIM + n];
    float* p = Bu + ((size_t)b * TSEQ + (size_t)c * CHUNKLEN) * NDIM + n;
#pragma unroll 4
    for (int t = 0; t < CHUNKLEN; ++t) {
        x = fmaf(abar, x, p[(size_t)t * NDIM]);
        p[(size_t)t * NDIM] = x;
    }
}

extern "C" void kernel_launch(void* const* d_in, const int* in_sizes, int n_in,
                              void* d_out, int out_size, void* d_ws, size_t ws_size,
                              hipStream_t stream) {
    const float* u      = (const float*)d_in[0];
    const float* log_A  = (const float*)d_in[1];
    const float* W_B    = (const float*)d_in[2];
    const float* b_B    = (const float*)d_in[3];
    const float* W_C    = (const float*)d_in[4];
    const float* b_C    = (const float*)d_in[5];
    const float* W_D    = (const float*)d_in[6];
    const float* b_D    = (const float*)d_in[7];
    const float* log_dt = (const float*)d_in[8];

    float* ws_x = (float*)d_ws;                              // Bu -> xs, 64 MB
    float* cs   = ws_x + (size_t)MTOT * NDIM;                // chunk states, 512 KB

    const dim3 blk(256);
    const dim3 gg(NDIM / 128, MTOT / 64);                    // (8, 256)

    // 1) Bu = dt * (u @ W_B^T + b_B)
    ssm_gemm_wmma<<<gg, blk, 0, stream>>>(u, W_B, b_B, nullptr, nullptr, nullptr,
                                          log_dt, ws_x);
    // 2-4) diagonal scan, chunk-parallel
    ssm_scan_pass1<<<(BATCH * NCHUNK * NDIM) / 256, blk, 0, stream>>>(ws_x, log_A, log_dt, cs);
    ssm_scan_pass2<<<(BATCH * NDIM) / 256, blk, 0, stream>>>(cs, log_A, log_dt);
    ssm_scan_pass3<<<(BATCH * NCHUNK * NDIM) / 256, blk, 0, stream>>>(ws_x, cs, log_A, log_dt);
    // 5) y = xs @ W_C^T + u @ W_D^T + (b_C + b_D)   (fused dual GEMM)
    ssm_gemm_wmma<<<gg, blk, 0, stream>>>(ws_x, W_C, b_C, u, W_D, b_D,
                                          nullptr, (float*)d_out);
}